// KPToSkl_65154653880889
// MI455X (gfx1250) — compile-verified
//
#include <hip/hip_runtime.h>

typedef float v2f __attribute__((ext_vector_type(2)));
typedef float v8f __attribute__((ext_vector_type(8)));

#define TS         224
#define NPIX       (TS * TS)        // 50176
#define TILES_IMG  (NPIX / 16)      // 3136
#define TPW        4                // pixel tiles per wave (64 pixels)
#define WAVES_IMG  (TILES_IMG / TPW) // 784
#define NBATCH     32
#define GAMMA_F    0.2f

// 17 real edges, padded to 32 with index 0 (dummy params substituted in-kernel)
__constant__ int c_pi[32] = {0,1,2,3,1,5,6,1,8,9,1,11,12,0,14,0,16,
                             0,0,0,0,0,0,0,0,0,0,0,0,0,0,0};
__constant__ int c_pj[32] = {1,2,3,4,5,6,7,8,9,10,11,12,13,14,15,16,17,
                             0,0,0,0,0,0,0,0,0,0,0,0,0,0,0};

__global__ __launch_bounds__(256) void kps_skl_heatmap(const float* __restrict__ kps,
                                                       float* __restrict__ out) {
    const int lane = threadIdx.x & 31;
    const int wave = blockIdx.x * (blockDim.x >> 5) + (threadIdx.x >> 5);
    const int b     = wave / WAVES_IMG;
    const int base0 = (wave % WAVES_IMG) * (16 * TPW);

    const int  eh = lane & 15;     // edge-in-tile for B/D, pixel-in-tile for A
    const bool hi = lane >= 16;

    // ---------------- per-lane edge parameters for both 16-edge tiles (hoisted) -------
    const float* kb = kps + b * 18 * 2;
    float vns[2], ivn[2];
    v2f Buv[2], Buu[2];
#pragma unroll
    for (int tt = 0; tt < 2; ++tt) {
        const int  e    = eh + tt * 16;
        const bool real = (e < 17);
        const int  ii = c_pi[e], jj = c_pj[e];
        const float pix_ = kb[ii * 2 + 0], piy_ = kb[ii * 2 + 1];
        float pjx_ = kb[jj * 2 + 0], pjy_ = kb[jj * 2 + 1];
        float vx_ = real ? (pix_ - pjx_) : 1.0f;
        float vy_ = real ? (piy_ - pjy_) : 0.0f;
        pjx_ = real ? pjx_ : 1.0e3f;   // dummy edge: far away -> beta underflows to 0
        pjy_ = real ? pjy_ : 1.0e3f;
        vns[tt] = vx_ * vx_ + vy_ * vy_;
        ivn[tt] = __builtin_amdgcn_rcpf(vns[tt]);          // single v_rcp_f32
        const float cc = -(pjx_ * vx_ + pjy_ * vy_);       // -pj.v
        const float ss = pjx_ * pjx_ + pjy_ * pjy_;        // |pj|^2
        // B 4x16 layout mirror of A: lanes 0-15 hold K=0,1 ; lanes 16-31 hold K=2,3
        Buv[tt].x = hi ? 0.0f : vx_;
        Buv[tt].y = hi ? cc   : vy_;
        Buu[tt].x = hi ? 1.0f : (-2.0f * pjx_);
        Buu[tt].y = hi ? ss   : (-2.0f * pjy_);
    }

    const v8f   zero = {0.f, 0.f, 0.f, 0.f, 0.f, 0.f, 0.f, 0.f};
    const float invn = 1.0f / (float)(TS - 1);
    float*      ob   = out + b * NPIX;

#pragma unroll
    for (int it = 0; it < TPW; ++it) {
        const int base = base0 + it * 16;

        // ---- A matrix: 16 pixels x 4 features [gx, gy, gx^2+gy^2, 1] ----
        // 32-bit A 16x4 layout: lanes 0-15 hold K=0,1 ; lanes 16-31 hold K=2,3 (M=lane&15)
        const int   p  = base + eh;
        const int   gi = p / TS;
        const int   gj = p - gi * TS;
        const float gx = (float)gi * invn;
        const float gy = (float)gj * invn;
        const float q  = gx * gx + gy * gy;
        v2f A;
        A.x = hi ? q    : gx;
        A.y = hi ? 1.0f : gy;

        // D[m,e]: u.v and |u|^2 for 16 pixels x 16 edges per WMMA
        v8f uv0 = __builtin_amdgcn_wmma_f32_16x16x4_f32(false, A, false, Buv[0], (short)0, zero, false, false);
        v8f uu0 = __builtin_amdgcn_wmma_f32_16x16x4_f32(false, A, false, Buu[0], (short)0, zero, false, false);
        v8f uv1 = __builtin_amdgcn_wmma_f32_16x16x4_f32(false, A, false, Buv[1], (short)0, zero, false, false);
        v8f uu1 = __builtin_amdgcn_wmma_f32_16x16x4_f32(false, A, false, Buu[1], (short)0, zero, false, false);

        // ---- epilogue: r, d^2, exp, max over both edge tiles ----
        // D layout: lane holds edge N = lane&15; pixels M = 8*(lane>=16) + vgpr(0..7)
        float mx[8];
#pragma unroll
        for (int m = 0; m < 8; ++m) {
            float u0 = uv0[m], w0 = uu0[m];
            float r0 = fminf(fmaxf(u0 * ivn[0], 0.0f), 1.0f);   // v_max + v_min (clip)
            float d0 = fmaf(r0 * r0, vns[0], fmaf(-2.0f * r0, u0, w0));
            float beta = __expf(-GAMMA_F * d0);

            float u1 = uv1[m], w1 = uu1[m];
            float r1 = fminf(fmaxf(u1 * ivn[1], 0.0f), 1.0f);
            float d1 = fmaf(r1 * r1, vns[1], fmaf(-2.0f * r1, u1, w1));
            beta = fmaxf(beta, __expf(-GAMMA_F * d1));
            mx[m] = beta;
        }

        // max across the 16 edge lanes (xor stays inside each 16-lane half)
#pragma unroll
        for (int s = 1; s < 16; s <<= 1) {
#pragma unroll
            for (int m = 0; m < 8; ++m)
                mx[m] = fmaxf(mx[m], __shfl_xor(mx[m], s, 32));
        }

        // lanes eh==0 of each half write 8 pixels (two float4, 64B-aligned base)
        if (eh == 0) {
            float4* o = (float4*)(ob + base + (hi ? 8 : 0));
            o[0] = make_float4(mx[0], mx[1], mx[2], mx[3]);
            o[1] = make_float4(mx[4], mx[5], mx[6], mx[7]);
        }
    }
}

extern "C" void kernel_launch(void* const* d_in, const int* in_sizes, int n_in,
                              void* d_out, int out_size, void* d_ws, size_t ws_size,
                              hipStream_t stream) {
    (void)in_sizes; (void)n_in; (void)d_ws; (void)ws_size; (void)out_size;
    const float* kps = (const float*)d_in[0];
    float*       out = (float*)d_out;

    const int total_waves = NBATCH * WAVES_IMG;   // 25088, exact multiple of 8
    const int blocks      = total_waves / 8;      // 3136 blocks of 256 threads (8 waves)
    kps_skl_heatmap<<<blocks, 256, 0, stream>>>(kps, out);
}